// GaussianMomentDescriptorT_5892695130818
// MI455X (gfx1250) — compile-verified
//
#include <hip/hip_runtime.h>
#include <math.h>

// ---------------- problem constants ----------------
#define N_ATOMS   10000
#define N_EDGES   500000
#define N_RADIAL  5
#define N_BASIS   7
#define N_SPECIES 119
#define R_MIN_F   0.5f
#define R_MAX_F   6.0f
#define PI_F      3.14159265358979f
#define EMB_NORM_F 0.3779644730092272f   // 1/sqrt(7)

// per-atom moment layout in workspace (200 floats):
//   M0: [0,5)   M1: [5,20)   M2: [20,65)   M3: [65,200)
#define M_STRIDE 200
// per-atom output layout (360 floats):
//   c0:0  c1:5  c2:20  c3:35  c4:50  c5:85  c6:160  c7:235
#define OUT_STRIDE 360

// padded per-wave LDS staging (floats): zero-filled so WMMA fragment reads
// are unconditional ds loads (no exec-mask predication).
//   A0 : 16x28  rows 0-4 = M3f(5x27), 5-9 = M2f(5x9), 10-14 = M1(5x3)
//   Y  : 16x12  Y[r*3+k][i*3+j] = M3[r,i,j,k]
//   U  : 16x4   U[s*3+i][j]     = M2[s,i,j]
//   B3 : 16x12  B3[n][k]        = M2f[n][k]        (B operand for H = Y*M2f^T)
//   M0 : 8
#define OFF_A0 0
#define OFF_Y  448
#define OFF_U  640
#define OFF_B3 704
#define OFF_M0 896
#define P_SIZE 904

typedef __attribute__((ext_vector_type(2))) float v2f;
typedef __attribute__((ext_vector_type(8))) float v8f;

#if defined(__has_builtin)
#if __has_builtin(__builtin_amdgcn_wmma_f32_16x16x4_f32)
#define GM_HAS_WMMA_F32 1
#endif
#endif
#ifndef GM_HAS_WMMA_F32
#define GM_HAS_WMMA_F32 0
#endif

// gfx1250 async global->LDS staging (ASYNCcnt-tracked). Flip to 0 to revert.
#if defined(__gfx1250__)
#define GM_USE_ASYNC 1
#else
#define GM_USE_ASYNC 0
#endif

// lower-triangle index tables (match reference tril_2d / tril_3d ordering)
__device__ __constant__ int TRI2I[15] = {0,1,1,2,2,2,3,3,3,3,4,4,4,4,4};
__device__ __constant__ int TRI2J[15] = {0,0,1,0,1,2,0,1,2,3,0,1,2,3,4};
__device__ __constant__ int TRI3I[35] = {0, 1,1,1, 2,2,2,2,2,2,
                                         3,3,3,3,3,3,3,3,3,3,
                                         4,4,4,4,4,4,4,4,4,4,4,4,4,4,4};
__device__ __constant__ int TRI3J[35] = {0, 0,1,1, 0,1,1,2,2,2,
                                         0,1,1,2,2,2,3,3,3,3,
                                         0,1,1,2,2,2,3,3,3,3,4,4,4,4,4};
__device__ __constant__ int TRI3K[35] = {0, 0,0,1, 0,0,1,0,1,2,
                                         0,0,1,0,1,2,0,1,2,3,
                                         0,0,1,0,1,2,0,1,2,3,0,1,2,3,4};

// ---------------- phase 0: zero the moment accumulator ----------------
__global__ void gm_zero(float* __restrict__ p, int n) {
  int i = blockIdx.x * blockDim.x + threadIdx.x;
  int stride = gridDim.x * blockDim.x;
  for (; i < n; i += stride) p[i] = 0.0f;
}

// ---------------- phase 1: per-edge moments, scatter-add by idx_j ----------------
__global__ __launch_bounds__(256) void gm_edges(
    const float* __restrict__ dr_vec,   // (N_EDGES,3)
    const int*   __restrict__ Z,        // (N_ATOMS)
    const int*   __restrict__ nbr,      // (2,N_EDGES)
    const float* __restrict__ emb,      // (119,119,5,7)
    float*       __restrict__ M)        // (N_ATOMS,200) accumulator
{
  int e = blockIdx.x * blockDim.x + threadIdx.x;
  if (e >= N_EDGES) return;

  int ii = nbr[e];
  int jj = nbr[N_EDGES + e];

  float x = dr_vec[3*e + 0];
  float y = dr_vec[3*e + 1];
  float z = dr_vec[3*e + 2];
  float dr  = sqrtf(x*x + y*y + z*z);
  float inv = 1.0f / (dr + 1e-5f);
  float dn[3] = {x*inv, y*inv, z*inv};

  const float betta    = (float)(N_BASIS*N_BASIS) / (R_MAX_F*R_MAX_F);
  const float rad_norm = powf(2.0f*betta/PI_F, 0.75f);

  float basis[N_BASIS];
#pragma unroll
  for (int b = 0; b < N_BASIS; ++b) {
    float shift = R_MIN_F + (R_MAX_F - R_MIN_F)/(float)N_BASIS * (float)b;
    float t = dr - shift;
    basis[b] = rad_norm * expf(-betta * t * t);
  }

  const float* c = emb + ((size_t)(Z[ii]*N_SPECIES + Z[jj])) * (N_RADIAL*N_BASIS);
  float cut = 0.5f * (cosf(PI_F * fminf(dr, R_MAX_F) / R_MAX_F) + 1.0f);

  float radial[N_RADIAL];
#pragma unroll
  for (int r = 0; r < N_RADIAL; ++r) {
    float acc = 0.0f;
#pragma unroll
    for (int b = 0; b < N_BASIS; ++b) acc += c[r*N_BASIS + b] * basis[b];
    radial[r] = acc * EMB_NORM_F * cut;
  }

  float* Ma = M + (size_t)jj * M_STRIDE;   // L2-resident accumulator
#pragma unroll
  for (int r = 0; r < N_RADIAL; ++r) {
    float m0 = radial[r];
    atomicAdd(&Ma[r], m0);
#pragma unroll
    for (int i = 0; i < 3; ++i) {
      float m1 = m0 * dn[i];
      atomicAdd(&Ma[5 + r*3 + i], m1);
#pragma unroll
      for (int j = 0; j < 3; ++j) {
        float m2 = m1 * dn[j];
        atomicAdd(&Ma[20 + r*9 + i*3 + j], m2);
#pragma unroll
        for (int k = 0; k < 3; ++k) {
          atomicAdd(&Ma[65 + r*27 + i*9 + j*3 + k], m2 * dn[k]);
        }
      }
    }
  }
}

// ---------------- phase 2 helpers ----------------
// D(16x16) += A(16xK) * B(Kx16) where B[k][n] is generated from column index n
// with the same per-lane formula as A rows (valid for Gram-style operands).
// Operands come from pre-zeroed padded LDS, so every read is unconditional.
template<int KSTR>
__device__ inline void gm_gram(const float* __restrict__ A,
                               const float* __restrict__ B,
                               float* __restrict__ g, int lane) {
#if GM_HAS_WMMA_F32
  const int row = lane & 15;
  const int kb  = (lane >> 4) << 1;   // K split: lanes 0-15 -> K+0/1, 16-31 -> K+2/3
  v8f acc = {0.f,0.f,0.f,0.f,0.f,0.f,0.f,0.f};
#pragma unroll
  for (int k0 = 0; k0 < KSTR; k0 += 4) {
    v2f a, b;
    a.x = A[row*KSTR + k0 + kb];
    a.y = A[row*KSTR + k0 + kb + 1];
    b.x = B[row*KSTR + k0 + kb];
    b.y = B[row*KSTR + k0 + kb + 1];
    acc = __builtin_amdgcn_wmma_f32_16x16x4_f32(
        false, a, false, b, (short)0, acc, false, false);
  }
  const int mhi = (lane >> 4) << 3;   // C/D: VGPR v -> M = v + 8*(lane>=16), N = lane&15
#pragma unroll
  for (int v = 0; v < 8; ++v) g[(mhi + v)*16 + row] = acc[v];
#else
  for (int idx = lane; idx < 256; idx += 32) {
    int m = idx >> 4, n = idx & 15;
    float s = 0.0f;
    for (int k = 0; k < KSTR; ++k) s += A[m*KSTR + k] * B[n*KSTR + k];
    g[idx] = s;
  }
#endif
}

// ---------------- phase 2: per-atom contractions (1 wave32 per atom) ----------------
__global__ __launch_bounds__(256) void gm_contract(
    const float* __restrict__ M, float* __restrict__ out)
{
  __shared__ float sP[8][P_SIZE];  // padded WMMA operands, per wave
  __shared__ float sG[8][256];     // raw moment stage, then 16x16 results
  int wave = threadIdx.x >> 5;
  int lane = threadIdx.x & 31;
  int atom = blockIdx.x * 8 + wave;
  if (atom >= N_ATOMS) atom = N_ATOMS - 1;   // uniform per wave; dup writes benign

  const float* Ma = M   + (size_t)atom * M_STRIDE;
  float*       Oa = out + (size_t)atom * OUT_STRIDE;
  float* P = sP[wave];
  float* g = sG[wave];

  // zero padded staging (overlapped with async global->LDS copy below)
  for (int t = lane; t < P_SIZE; t += 32) P[t] = 0.0f;

  // raw moments -> g[0..200)
#if GM_USE_ASYNC
  for (int t = lane; t < M_STRIDE; t += 32) {
    unsigned lds_addr = (unsigned)(size_t)(const void*)(g + t); // addr[31:0] = LDS byte addr
    unsigned goff     = (unsigned)(t * 4);
    asm volatile("global_load_async_to_lds_b32 %0, %1, %2"
                 :: "v"(lds_addr), "v"(goff), "s"(Ma) : "memory");
  }
  asm volatile("s_wait_asynccnt 0" ::: "memory");
#else
  for (int t = lane; t < M_STRIDE; t += 32) g[t] = Ma[t];
#endif

  // scatter raw moments into padded operand matrices (same-wave LDS is in-order)
  for (int t = lane; t < M_STRIDE; t += 32) {
    float v = g[t];
    if (t < 5) {
      P[OFF_M0 + t] = v;
    } else if (t < 20) {                       // M1[r][i]
      int q = t - 5;
      P[OFF_A0 + (10 + q/3)*28 + (q%3)] = v;
    } else if (t < 65) {                       // M2[r][i][j]
      int q = t - 20;
      P[OFF_A0 + (5 + q/9)*28 + (q%9)] = v;
      P[OFF_B3 + (q/9)*12 + (q%9)]     = v;
      P[OFF_U  + (q/3)*4  + (q%3)]     = v;
    } else {                                   // M3[r][i][j][k], q = r*27+i*9+j*3+k
      int q = t - 65;
      P[OFF_A0 + (q/27)*28 + (q%27)]                   = v;
      P[OFF_Y  + ((q/27)*3 + (q%3))*12 + ((q/3)%9)]    = v;
    }
  }
  __syncthreads();

  const float* A0  = P + OFF_A0;
  const float* Uu  = P + OFF_U;
  const float* Yy  = P + OFF_Y;
  const float* B3  = P + OFF_B3;
  const float* M0p = P + OFF_M0;

  // ---- big Gram (7 WMMAs, K=28): c3 = rows 0-4, c2 = 5-9, c1 = 10-14 ----
  gm_gram<28>(A0, A0, g, lane);
  __syncthreads();
  if (lane < 5) Oa[lane] = M0p[lane];                       // c0
  for (int p = lane; p < 15; p += 32) {
    int i = TRI2I[p], j = TRI2J[p];
    Oa[5  + p] = g[(10+i)*16 + (10+j)];                     // c1
    Oa[20 + p] = g[(5+i)*16  + (5+j)];                      // c2
    Oa[35 + p] = g[i*16 + j];                               // c3
  }
  __syncthreads();

  // ---- Q = U*U^T (1 WMMA): Q[(s,i),(t,j)] = sum_k M2[s,i,k] M2[t,j,k] ----
  gm_gram<4>(Uu, Uu, g, lane);
  __syncthreads();
  for (int o = lane; o < 35; o += 32) {                     // c4
    int r = TRI3I[o], s = TRI3J[o], t = TRI3K[o];
    float a = 0.0f;
#pragma unroll
    for (int i = 0; i < 3; ++i)
#pragma unroll
      for (int j = 0; j < 3; ++j)
        a += A0[(5+r)*28 + i*3 + j] * g[(s*3+i)*16 + (t*3+j)];
    Oa[50 + o] = a;
  }
  for (int o = lane; o < 75; o += 32) {                     // c5 (pure VALU)
    int p = o / 5, t = o % 5;
    int r = TRI2I[p], s = TRI2J[p];
    float a = 0.0f;
#pragma unroll
    for (int i = 0; i < 3; ++i)
#pragma unroll
      for (int j = 0; j < 3; ++j)
        a += A0[(10+r)*28 + i] * A0[(10+s)*28 + j] * A0[(5+t)*28 + i*3 + j];
    Oa[85 + o] = a;
  }
  __syncthreads();

  // ---- G6 = Y*Y^T (3 WMMAs): G6[(r,k),(s,l)] = sum_ij M3[r,i,j,k] M3[s,i,j,l] ----
  gm_gram<12>(Yy, Yy, g, lane);
  __syncthreads();
  for (int o = lane; o < 75; o += 32) {                     // c6
    int p = o / 5, t = o % 5;
    int r = TRI2I[p], s = TRI2J[p];
    float a = 0.0f;
#pragma unroll
    for (int k = 0; k < 3; ++k)
#pragma unroll
      for (int l = 0; l < 3; ++l)
        a += g[(r*3+k)*16 + (s*3+l)] * A0[(5+t)*28 + k*3 + l];
    Oa[160 + o] = a;
  }
  __syncthreads();

  // ---- H = Y*M2f^T (3 WMMAs): H[(r,k),s] = sum_ij M3[r,i,j,k] M2[s,i,j] ----
  gm_gram<12>(Yy, B3, g, lane);
  __syncthreads();
  for (int o = lane; o < 125; o += 32) {                    // c7 (full 5x5x5)
    int r = o / 25, s = (o / 5) % 5, t = o % 5;
    float a = 0.0f;
#pragma unroll
    for (int k = 0; k < 3; ++k)
      a += g[(r*3+k)*16 + s] * A0[(10+t)*28 + k];
    Oa[235 + o] = a;
  }
}

// ---------------- launcher ----------------
extern "C" void kernel_launch(void* const* d_in, const int* in_sizes, int n_in,
                              void* d_out, int out_size, void* d_ws, size_t ws_size,
                              hipStream_t stream) {
  const float* dr_vec = (const float*)d_in[0];
  const int*   Z      = (const int*)d_in[1];
  const int*   nbr    = (const int*)d_in[2];
  const float* emb    = (const float*)d_in[3];
  float* out = (float*)d_out;
  float* M   = (float*)d_ws;                 // N_ATOMS*200 floats = 8 MB scratch

  const int nM = N_ATOMS * M_STRIDE;
  hipLaunchKernelGGL(gm_zero, dim3(512), dim3(256), 0, stream, M, nM);
  hipLaunchKernelGGL(gm_edges, dim3((N_EDGES + 255) / 256), dim3(256), 0, stream,
                     dr_vec, Z, nbr, emb, M);
  hipLaunchKernelGGL(gm_contract, dim3((N_ATOMS + 7) / 8), dim3(256), 0, stream,
                     M, out);
}